// GCN_25872882991698
// MI455X (gfx1250) — compile-verified
//
#include <hip/hip_runtime.h>
#include <stdint.h>

typedef __attribute__((ext_vector_type(2))) float v2f;
typedef __attribute__((ext_vector_type(8))) float v8f;

#define GCN_N_FEAT 256
#define GCN_OUT    128
#define W_LDS_STRIDE 260          // 256 + 4 pad -> stride mod 64 banks = 4 (conflict-free)
#define W_LDS_BYTES  (GCN_OUT * W_LDS_STRIDE * 4)   // 133120

// ---------------------------------------------------------------------------
// 1) degree init: every node has a self-loop -> deg starts at 1
// ---------------------------------------------------------------------------
__global__ void gcn_deg_init(float* __restrict__ deg, int N) {
  int i = blockIdx.x * blockDim.x + threadIdx.x;
  if (i < N) deg[i] = 1.0f;
}

// 2) per-edge degree count over the target (col) index
__global__ void gcn_deg_count(const int* __restrict__ col, float* __restrict__ deg, int E) {
  int e = blockIdx.x * blockDim.x + threadIdx.x;
  if (e < E) atomicAdd(&deg[col[e]], 1.0f);
}

// 3) dinv = rsqrt(deg) in place
__global__ void gcn_dinv(float* __restrict__ deg, int N) {
  int i = blockIdx.x * blockDim.x + threadIdx.x;
  if (i < N) {
    float d = deg[i];
    deg[i] = d > 0.0f ? rsqrtf(d) : 0.0f;
  }
}

// ---------------------------------------------------------------------------
// 4) h = x @ W^T  via V_WMMA_F32_16X16X4_F32  (fp32 in, fp32 out)
//    - W (128x256 fp32 = 128KB) is staged once per workgroup into LDS with
//      GLOBAL_LOAD_ASYNC_TO_LDS_B128 (ASYNCcnt), padded to 260-float rows so
//      the 16-lane B-fragment column reads are LDS bank-conflict-free.
//    - 8 waves/block, each wave owns a 16-row strip of x and all 8 OUT tiles.
//    A 16x4 layout : lane holds row m0+(lane&15), K pair = 2*(lane>>4)
//    B  4x16 layout: B[k][n] = W[n][k] -> lane holds col n0+(lane&15)
//    C/D layout    : VGPR j -> M = j + 8*(lane>>4), N = lane&15
// ---------------------------------------------------------------------------
__global__ void __launch_bounds__(256)
gcn_gemm_xwt(const float* __restrict__ x, const float* __restrict__ W,
             float* __restrict__ h, int N) {
  extern __shared__ float wlds[];    // [GCN_OUT][W_LDS_STRIDE]

  const int tid = threadIdx.x;

  // ---- async copy W -> LDS (all 256 threads, before any wave exits) ----
  {
    // LDS byte address = low 32 bits of the generic address (ISA 10.2 aperture map)
    uint32_t ldsbase = (uint32_t)(uintptr_t)(void*)wlds;
#pragma unroll
    for (int it = 0; it < (GCN_OUT * GCN_N_FEAT * 4) / (256 * 16); ++it) {  // 32 iters
      int g   = it * 256 + tid;          // 16B chunk index
      int r   = g >> 6;                  // W row (64 chunks of 16B per 1KB row)
      int ci  = g & 63;                  // chunk within row
      uint32_t ldsaddr = ldsbase + (uint32_t)(r * (W_LDS_STRIDE * 4) + ci * 16);
      int goff = g * 16;                 // byte offset into W
      asm volatile("global_load_async_to_lds_b128 %0, %1, %2"
                   :: "v"(ldsaddr), "v"(goff), "s"(W)
                   : "memory");
    }
    asm volatile("s_wait_asynccnt 0" ::: "memory");
  }
  __syncthreads();

  const int lane = tid & 31;
  const int wave = tid >> 5;
  const int m0   = (blockIdx.x * 8 + wave) * 16;
  if (m0 >= N) return;                   // wave-uniform exit (after the barrier)

  const int half = lane >> 4;            // 0 | 1
  const int l16  = lane & 15;
  int mrow = m0 + l16;
  if (mrow >= N) mrow = N - 1;           // clamp (N is 16-aligned here, safe)
  const float* xrow = x + (size_t)mrow * GCN_N_FEAT + 2 * half;

  const float* bptr[8];
#pragma unroll
  for (int t = 0; t < 8; ++t)
    bptr[t] = wlds + (t * 16 + l16) * W_LDS_STRIDE + 2 * half;

  v8f acc[8];
#pragma unroll
  for (int t = 0; t < 8; ++t) acc[t] = (v8f)0.0f;

#pragma unroll 4
  for (int k0 = 0; k0 < GCN_N_FEAT; k0 += 4) {
    v2f a = *(const v2f*)(xrow + k0);
#pragma unroll
    for (int t = 0; t < 8; ++t) {
      v2f b = *(const v2f*)(bptr[t] + k0);          // ds_load_b64, conflict-free
      acc[t] = __builtin_amdgcn_wmma_f32_16x16x4_f32(
          /*neg_a=*/false, a, /*neg_b=*/false, b,
          /*c_mod=*/(short)0, acc[t], /*reuse_a=*/false, /*reuse_b=*/false);
    }
  }

#pragma unroll
  for (int t = 0; t < 8; ++t) {
#pragma unroll
    for (int j = 0; j < 8; ++j) {
      int m = m0 + j + 8 * half;
      if (m < N) h[(size_t)m * GCN_OUT + t * 16 + l16] = acc[t][j];
    }
  }
}

// ---------------------------------------------------------------------------
// 5) out = bias + h * dinv^2   (self-loop message; norm = dinv[n]^2)
// ---------------------------------------------------------------------------
__global__ void gcn_self_init(const float* __restrict__ h, const float* __restrict__ dinv,
                              const float* __restrict__ bias, float* __restrict__ out,
                              int total) {
  int i = blockIdx.x * blockDim.x + threadIdx.x;
  if (i < total) {
    int n = i >> 7;          // /128
    int o = i & 127;
    float di = dinv[n];
    out[i] = bias[o] + h[i] * di * di;
  }
}

// ---------------------------------------------------------------------------
// 6) edge scatter: one wave per edge; scalar (readfirstlane) indices,
//    float4 gather from h[row], 4 fp32 atomic adds into out[col].
//    h (51MB) and out (51MB) are L2-resident (192MB) -> runs at L2 rate.
// ---------------------------------------------------------------------------
__global__ void __launch_bounds__(256)
gcn_edge_scatter(const int* __restrict__ row, const int* __restrict__ col,
                 const float* __restrict__ h, const float* __restrict__ dinv,
                 float* __restrict__ out, int E) {
  int lane = threadIdx.x & 31;
  int wid  = (blockIdx.x * blockDim.x + threadIdx.x) >> 5;
  if (wid >= E) return;                      // wave-uniform exit

  int r = __builtin_amdgcn_readfirstlane(row[wid]);
  int c = __builtin_amdgcn_readfirstlane(col[wid]);
  float norm = dinv[r] * dinv[c];

  const float4* hr = (const float4*)(h + (size_t)r * GCN_OUT);
  float4 v = hr[lane];
  float* dst = out + (size_t)c * GCN_OUT + lane * 4;
  atomicAdd(dst + 0, v.x * norm);
  atomicAdd(dst + 1, v.y * norm);
  atomicAdd(dst + 2, v.z * norm);
  atomicAdd(dst + 3, v.w * norm);
}

// ---------------------------------------------------------------------------
// 7) per-channel PReLU, in place (out fully rewritten each launch -> safe)
// ---------------------------------------------------------------------------
__global__ void gcn_prelu(float* __restrict__ out, const float* __restrict__ a, int total) {
  int i = blockIdx.x * blockDim.x + threadIdx.x;
  if (i < total) {
    float v  = out[i];
    float al = a[i & 127];
    out[i] = v > 0.0f ? v : al * v;
  }
}

// ---------------------------------------------------------------------------
extern "C" void kernel_launch(void* const* d_in, const int* in_sizes, int n_in,
                              void* d_out, int out_size, void* d_ws, size_t ws_size,
                              hipStream_t stream) {
  const float* x    = (const float*)d_in[0];
  const int*   ei   = (const int*)d_in[1];   // [2, E] int32 (JAX x64 off)
  const float* W    = (const float*)d_in[2];
  const float* bias = (const float*)d_in[3];
  const float* pa   = (const float*)d_in[4];
  float* out = (float*)d_out;

  const int N = in_sizes[0] / GCN_N_FEAT;    // 100000
  const int E = in_sizes[1] / 2;             // 3200000
  const int total = N * GCN_OUT;             // 12.8M

  const int* row = ei;         // edge_index[0]
  const int* col = ei + E;     // edge_index[1]

  // workspace: h [N,128] fp32, then deg/dinv [N] fp32  (~51.6 MB)
  float* h    = (float*)d_ws;
  float* dinv = h + (size_t)N * GCN_OUT;

  // degree / normalization
  gcn_deg_init <<<(N + 255) / 256, 256, 0, stream>>>(dinv, N);
  gcn_deg_count<<<(E + 255) / 256, 256, 0, stream>>>(col, dinv, E);
  gcn_dinv     <<<(N + 255) / 256, 256, 0, stream>>>(dinv, N);

  // dense linear via fp32 WMMA with async-LDS-staged W
  int strips = (N + 15) / 16;
  gcn_gemm_xwt<<<(strips + 7) / 8, 256, W_LDS_BYTES, stream>>>(x, W, h, N);

  // self-loop + bias, edge scatter-add, PReLU
  gcn_self_init   <<<(total + 255) / 256, 256, 0, stream>>>(h, dinv, bias, out, total);
  gcn_edge_scatter<<<(E + 7) / 8, 256, 0, stream>>>(row, col, h, dinv, out, E);
  gcn_prelu       <<<(total + 255) / 256, 256, 0, stream>>>(out, pa, total);
}